// LocalizationModule_55370718380390
// MI455X (gfx1250) — compile-verified
//
#include <hip/hip_runtime.h>
#include <hip/hip_bf16.h>
#include <math.h>

typedef __attribute__((ext_vector_type(2))) float v2f;
typedef __attribute__((ext_vector_type(8))) float v8f;

#define BATCH   16
#define HH      768
#define WW      768
#define NPIX    (BATCH * HH * WW)          // 9437184
#define NBLK    (NPIX / 256)               // 36864 (each block = 256 consecutive flat pixels)
#define MAX_DET 393216
#define NEGBIG  (-3.0e38f)

// ---------------------------------------------------------------------------
// Kernel 1: 5x5 strict local-max detection (LDS-tiled stencil).
// Each block covers 256 consecutive flat pixels (always inside one image row:
// W=768 divisible by 256). Emits 8x32-bit mask words + a count per block.
// ---------------------------------------------------------------------------
__global__ void lm_detect_kernel(const float* __restrict__ x,
                                 unsigned int* __restrict__ masks,
                                 unsigned int* __restrict__ counts) {
  __shared__ float tile[5][260];
  __shared__ unsigned int mwords[8];

  const int tid = threadIdx.x;
  const long p0 = (long)blockIdx.x * 256;
  const int b  = (int)(p0 / (HH * WW));
  const int rc = (int)(p0 % (HH * WW));
  const int r  = rc / WW;
  const int c0 = rc % WW;
  const float* heat = x + ((size_t)(2 * b + 1)) * (HH * WW);

  if (tid < 8) mwords[tid] = 0u;

  // Stage 5 rows x 260 cols of thresholded values; out-of-image = -inf (pad).
  for (int i = tid; i < 5 * 260; i += 256) {
    const int dy = i / 260;
    const int dx = i % 260;
    const int rr = r + dy - 2;
    const int cc = c0 + dx - 2;
    float v = NEGBIG;
    if (rr >= 0 && rr < HH && cc >= 0 && cc < WW) {
      const float h = heat[rr * WW + cc];
      v = (h > 0.5f) ? h : 0.0f;
    }
    tile[dy][dx] = v;
  }
  __syncthreads();

  const int lc = tid + 2;
  const float thc = tile[2][lc];
  float m = NEGBIG;
#pragma unroll
  for (int dy = 0; dy < 5; ++dy) {
#pragma unroll
    for (int dx = 0; dx < 5; ++dx) {
      if (dy == 2 && dx == 2) continue;
      m = fmaxf(m, tile[dy][lc - 2 + dx]);
    }
  }
  const bool isMax = thc > m;
  if (isMax) atomicOr(&mwords[tid >> 5], 1u << (tid & 31));
  __syncthreads();

  if (tid < 8) masks[(size_t)blockIdx.x * 8 + tid] = mwords[tid];
  if (tid == 0) {
    unsigned int s = 0;
#pragma unroll
    for (int j = 0; j < 8; ++j) s += __popc(mwords[j]);
    counts[blockIdx.x] = s;
  }
}

// ---------------------------------------------------------------------------
// Kernel 2: single-block exclusive scan of NBLK counts (36 chunks of 1024).
// ---------------------------------------------------------------------------
__global__ void lm_scan_kernel(const unsigned int* __restrict__ counts,
                               unsigned int* __restrict__ offsets,
                               unsigned int* __restrict__ total) {
  __shared__ unsigned int s[1024];
  __shared__ unsigned int carry;
  const int tid = threadIdx.x;
  if (tid == 0) carry = 0;
  __syncthreads();

  for (int base = 0; base < NBLK; base += 1024) {
    const int i = base + tid;
    const unsigned int v = (i < NBLK) ? counts[i] : 0u;
    s[tid] = v;
    __syncthreads();
    // Hillis-Steele inclusive scan
    for (int off = 1; off < 1024; off <<= 1) {
      const unsigned int t = (tid >= off) ? s[tid - off] : 0u;
      __syncthreads();
      s[tid] += t;
      __syncthreads();
    }
    if (i < NBLK) offsets[i] = carry + s[tid] - v;  // exclusive
    __syncthreads();
    if (tid == 1023) carry += s[1023];
    __syncthreads();
  }
  if (tid == 0) *total = carry;
}

// ---------------------------------------------------------------------------
// Kernel 3: ordered compaction of flat indices (matches jnp.nonzero order).
// ---------------------------------------------------------------------------
__global__ void lm_compact_kernel(const unsigned int* __restrict__ masks,
                                  const unsigned int* __restrict__ offsets,
                                  unsigned int* __restrict__ flatIdx) {
  const int tid = threadIdx.x;
  const int word = tid >> 5;
  const int lane = tid & 31;
  const unsigned int w = masks[(size_t)blockIdx.x * 8 + word];
  if ((w >> lane) & 1u) {
    unsigned int before = 0;
    for (int j = 0; j < word; ++j) before += __popc(masks[(size_t)blockIdx.x * 8 + j]);
    const unsigned int inword = __popc(w & ((1u << lane) - 1u));
    const unsigned int pos = offsets[blockIdx.x] + before + inword;
    if (pos < (unsigned int)MAX_DET)
      flatIdx[pos] = (unsigned int)(blockIdx.x * 256 + tid);
  }
}

// Branchless basis evaluation: [1, x, x^2, y, x*y, y^2]
__device__ __forceinline__ float basisf(int m, float xx, float yy) {
  float r = 0.0f;
  r = (m == 0) ? 1.0f    : r;
  r = (m == 1) ? xx      : r;
  r = (m == 2) ? xx * xx : r;
  r = (m == 3) ? yy      : r;
  r = (m == 4) ? xx * yy : r;
  r = (m == 5) ? yy * yy : r;
  return r;
}

// ---------------------------------------------------------------------------
// Kernel 4: one wave32 per detection.
//   A (16x4 f32) = Aw^T chunk (rows = 6 weighted-basis features, K = samples)
//   B (4x16 f32) = [Aw | bvec] chunk (cols 0..5 = Aw, col 6 = w*log(z))
//   7 chained V_WMMA_F32_16X16X4_F32 cover k=0..27 (25 real samples).
//   D[0:6,0:6] = ATA, D[0:6,6] = ATb  -> lane n holds column n in acc[0..5].
//   Solve 6x6 by column-parallel Gaussian elimination via wave shuffles.
// ---------------------------------------------------------------------------
__global__ void lm_fit_kernel(const float* __restrict__ x,
                              const unsigned int* __restrict__ flatIdx,
                              const unsigned int* __restrict__ totalp,
                              float* __restrict__ out) {
  const int tid = threadIdx.x;
  const int lane = tid & 31;
  const int d = blockIdx.x * 8 + (tid >> 5);   // grid sized so d < MAX_DET always

  const unsigned int tot = *totalp;
  const unsigned int usable = tot < (unsigned int)MAX_DET ? tot : (unsigned int)MAX_DET;
  const bool valid = (unsigned int)d < tot;
  const unsigned int flatp = ((unsigned int)d < usable) ? flatIdx[d] : 0u;

  const int b  = (int)(flatp / (HH * WW));
  const int rc = (int)(flatp % (HH * WW));
  const int r  = rc / WW;
  const int c  = rc % WW;
  const float* heat = x + ((size_t)(2 * b + 1)) * (HH * WW);

  // Lane k (0..24) owns sample k of the 5x5 patch (clipped coords, z clamp).
  const int k  = lane < 25 ? lane : 24;
  const int dy = k / 5 - 2;
  const int dx = k % 5 - 2;
  const int rr = min(max(r + dy, 0), HH - 1);
  const int cc = min(max(c + dx, 0), WW - 1);
  float z = heat[rr * WW + cc];
  z = fmaxf(z, 1e-6f);
  const float wgt = z * z;
  const float lzw = logf(z) * wgt;

  // Build normal equations with f32 WMMA: acc = sum_k A^T(:,k) * [Aw|b](k,:)
  v8f acc = {};
  const int half2 = (lane >> 4) << 1;  // K sub-pair handled by this half-wave
  const int mn = lane & 15;            // row (A) / col (B) index
#pragma unroll
  for (int k0 = 0; k0 < 28; k0 += 4) {
    const int ka = k0 + half2;
    const int kb = ka + 1;
    const float wA = __shfl(wgt, ka & 31, 32);
    const float wB = __shfl(wgt, kb & 31, 32);
    const float lA = __shfl(lzw, ka & 31, 32);
    const float lB = __shfl(lzw, kb & 31, 32);
    const float xA = (float)(ka % 5 - 2), yA = (float)(ka / 5 - 2);
    const float xB = (float)(kb % 5 - 2), yB = (float)(kb / 5 - 2);
    const bool okA = ka < 25, okB = kb < 25;
    const float fA = basisf(mn, xA, yA);
    const float fB = basisf(mn, xB, yB);
    const float a0 = (okA && mn < 6) ? fA * wA : 0.0f;
    const float a1 = (okB && mn < 6) ? fB * wB : 0.0f;
    const float b0 = okA ? ((mn < 6) ? fA * wA : ((mn == 6) ? lA : 0.0f)) : 0.0f;
    const float b1 = okB ? ((mn < 6) ? fB * wB : ((mn == 6) ? lB : 0.0f)) : 0.0f;
    v2f av; av.x = a0; av.y = a1;
    v2f bv; bv.x = b0; bv.y = b1;
    acc = __builtin_amdgcn_wmma_f32_16x16x4_f32(false, av, false, bv,
                                                (short)0, acc, false, false);
  }

  // Lane n (0..6) holds column n of [ATA | ATb] in acc[0..5].
  float col[6];
#pragma unroll
  for (int i = 0; i < 6; ++i) col[i] = acc[i];

  // Forward elimination (columns in lanes; multipliers broadcast from lane p).
#pragma unroll
  for (int p = 0; p < 5; ++p) {
    const float piv = __shfl(col[p], p, 32);
#pragma unroll
    for (int i = 0; i < 6; ++i) {
      if (i > p) {
        const float mfac = __shfl(col[i], p, 32) / piv;  // A[i][p]/A[p][p]
        col[i] -= mfac * col[p];
      }
    }
  }
  // Back substitution; every lane computes all 6 coefficients.
  float bb[6];
#pragma unroll
  for (int i = 0; i < 6; ++i) bb[i] = __shfl(col[i], 6, 32);
  float xc[6];
#pragma unroll
  for (int j = 5; j >= 0; --j) {
    const float Ujj = __shfl(col[j], j, 32);
    xc[j] = bb[j] / Ujj;
#pragma unroll
    for (int i = 0; i < 6; ++i)
      if (i < j) bb[i] -= xc[j] * __shfl(col[i], j, 32);
  }

  if (lane == 0) {
    const float Aq0 = xc[2], Aq1 = xc[5];
    const float Bl0 = xc[1], Bl1 = xc[3];
    const float Cc0 = xc[0], Cc1 = xc[4];
    const bool good = valid && (Aq0 < 0.0f) && (Aq1 < 0.0f);
    const float A0 = good ? Aq0 : -0.5f;
    const float A1 = good ? Aq1 : -0.5f;
    const float s0 = sqrtf(-1.0f / (2.0f * A0));
    const float s1 = sqrtf(-1.0f / (2.0f * A1));
    const float mu0 = Bl0 * s0 * s0;
    const float mu1 = Bl1 * s1 * s1;
    const float h0 = expf(good ? (Cc0 + 0.5f * mu0 * mu0 / (s0 * s0)) : 0.0f);
    const float h1 = expf(good ? (Cc1 + 0.5f * mu1 * mu1 / (s1 * s1)) : 0.0f);
    const float p0 = (float)r + mu1;   // pos = [r,c] + mu[::-1]
    const float p1 = (float)c + mu0;

    float* outS = out;
    float* outP = out + 2 * (size_t)MAX_DET;
    float* outH = out + 4 * (size_t)MAX_DET;
    float* outG = out + 6 * (size_t)MAX_DET;
    outS[2 * d]     = good ? s0 : 0.0f;
    outS[2 * d + 1] = good ? s1 : 0.0f;
    outP[2 * d]     = good ? p0 : 0.0f;
    outP[2 * d + 1] = good ? p1 : 0.0f;
    outH[2 * d]     = good ? h0 : 0.0f;
    outH[2 * d + 1] = good ? h1 : 0.0f;
    outG[d]         = good ? 1.0f : 0.0f;
  }
}

extern "C" void kernel_launch(void* const* d_in, const int* in_sizes, int n_in,
                              void* d_out, int out_size, void* d_ws, size_t ws_size,
                              hipStream_t stream) {
  (void)in_sizes; (void)n_in; (void)out_size; (void)ws_size;
  const float* x = (const float*)d_in[0];
  float* out = (float*)d_out;

  // Workspace layout (u32 words): masks | counts | offsets | total | flatIdx
  unsigned int* ws      = (unsigned int*)d_ws;
  unsigned int* masks   = ws;                        // NBLK*8
  unsigned int* counts  = masks + (size_t)NBLK * 8;  // NBLK
  unsigned int* offsets = counts + NBLK;             // NBLK
  unsigned int* total   = offsets + NBLK;            // 1
  unsigned int* flatIdx = total + 1;                 // MAX_DET

  lm_detect_kernel <<<NBLK, 256, 0, stream>>>(x, masks, counts);
  lm_scan_kernel   <<<1, 1024, 0, stream>>>(counts, offsets, total);
  lm_compact_kernel<<<NBLK, 256, 0, stream>>>(masks, offsets, flatIdx);
  lm_fit_kernel    <<<MAX_DET / 8, 256, 0, stream>>>(x, flatIdx, total, out);
}